// GatherModel_84542136254724
// MI455X (gfx1250) — compile-verified
//
#include <hip/hip_runtime.h>

#define DD   42      // hidden dim
#define DR   48      // padded node row stride (floats)
#define HR   44      // h row stride: 42 values + 1.0(bias col) + 0.0(pad col)
#define GJ   44      // K-group stride: c = i*44 + j, j in [0,43]
#define KPAD 1848    // 42*44, multiple of 4; 4-blocks never straddle i-groups
#define KC   256     // K-chunk staged in LDS
#define WGE  256     // edges per workgroup

typedef float v2f __attribute__((ext_vector_type(2)));
typedef float v8f __attribute__((ext_vector_type(8)));

__global__ void k_zero(float* __restrict__ p, int n) {
  int i = blockIdx.x * blockDim.x + threadIdx.x;
  if (i < n) p[i] = 0.f;
}

__global__ void k_indeg(const int* __restrict__ dst, int* __restrict__ cnt, int E) {
  int e = blockIdx.x * blockDim.x + threadIdx.x;
  if (e < E) atomicAdd(&cnt[dst[e]], 1);
}

__global__ void k_inv(const int* __restrict__ cnt, float* __restrict__ inv, int N) {
  int n = blockIdx.x * blockDim.x + threadIdx.x;
  if (n < N) inv[n] = 1.f / fmaxf((float)cnt[n], 1.f);
}

// out[n, 0..47] = relu(n_feat @ lin0_w^T + b), cols 42..47 zero
__global__ void k_lin0(const float* __restrict__ nf, const float* __restrict__ w,
                       const float* __restrict__ b, float* __restrict__ out, int N) {
  int idx = blockIdx.x * blockDim.x + threadIdx.x;
  int n = idx / DR, o = idx % DR;
  if (n >= N) return;
  float v = 0.f;
  if (o < DD) {
    float s = b[o];
    #pragma unroll
    for (int k = 0; k < DD; ++k) s += nf[n * DD + k] * w[o * DD + k];
    v = fmaxf(s, 0.f);
  }
  out[n * DR + o] = v;
}

// h[e, j] = LN(relu(e_feat @ en1_w^T + en1_b)) * g + b ; h[e,42]=1 ; h[e,43]=0
__global__ void k_edge_mlp(const float* __restrict__ ef, const float* __restrict__ w1,
                           const float* __restrict__ b1, const float* __restrict__ g,
                           const float* __restrict__ bb, float* __restrict__ h, int E) {
  int e = blockIdx.x * blockDim.x + threadIdx.x;
  if (e >= E) return;
  float x[10];
  #pragma unroll
  for (int k = 0; k < 10; ++k) x[k] = ef[e * 10 + k];
  float t[DD];
  float mu = 0.f;
  #pragma unroll
  for (int o = 0; o < DD; ++o) {
    float s = b1[o];
    #pragma unroll
    for (int k = 0; k < 10; ++k) s += x[k] * w1[o * 10 + k];
    s = fmaxf(s, 0.f);
    t[o] = s;
    mu += s;
  }
  mu *= (1.f / DD);
  float var = 0.f;
  #pragma unroll
  for (int o = 0; o < DD; ++o) { float d = t[o] - mu; var += d * d; }
  var *= (1.f / DD);
  float r = rsqrtf(var + 1e-5f);
  #pragma unroll
  for (int o = 0; o < DD; ++o) h[e * HR + o] = (t[o] - mu) * r * g[o] + bb[o];
  h[e * HR + 42] = 1.f;
  h[e * HR + 43] = 0.f;
}

// Wmat element (c, o) with c = i*44 + j:
//   j<42 -> en2_w[(i*42+o)*42 + j], j==42 -> en2_b[i*42+o], j==43 / o>=42 -> 0
// Stored K-BLOCKED: wm[((c>>2)*48 + o)*4 + (c&3)] so a column's 4 K-values are
// contiguous -> inner loop B-fragments load as one ds_load_b64.
__global__ void k_build_wmat(const float* __restrict__ w2, const float* __restrict__ b2,
                             float* __restrict__ wm) {
  int idx = blockIdx.x * blockDim.x + threadIdx.x;
  if (idx >= KPAD * DR) return;
  int c = idx / DR, o = idx % DR;
  int i = c / GJ, j = c - i * GJ;
  float v = 0.f;
  if (o < DD) {
    if (j < DD)       v = w2[(i * DD + o) * DD + j];
    else if (j == DD) v = b2[i * DD + o];
  }
  wm[((c >> 2) * DR + o) * 4 + (c & 3)] = v;
}

// Per step: msg = U' @ Wmat where U'[e, i*44+j] = x[e,i]*hh[e,j]; scatter into neigh
__global__ __launch_bounds__(256)
void k_msg_gemm(const float* __restrict__ out_cur, const float* __restrict__ h,
                const float* __restrict__ wm, const int* __restrict__ src,
                const int* __restrict__ dst, float* __restrict__ neigh, int E) {
  __shared__ float Wl[KC * DR];    // 49152 B (K-blocked: [KC/4][48][4])
  __shared__ float Xl[WGE * HR];   // 45056 B
  __shared__ float Hl[WGE * HR];   // 45056 B
  __shared__ int   Dl[WGE];

  const int tid   = threadIdx.x;
  const int wave  = tid >> 5;
  const int lane  = tid & 31;
  const int lhalf = lane >> 4;   // 0: lanes 0-15, 1: lanes 16-31
  const int lm    = lane & 15;
  const int ebase = blockIdx.x * WGE;

  // stage gathered x rows, hh rows, dst for this WG's 256 edges
  {
    int eg = ebase + tid;
    int ec = eg < E ? eg : (E - 1);
    int sn = src[ec];
    Dl[tid] = dst[ec];
    #pragma unroll
    for (int i = 0; i < HR; i += 4) {
      *(float4*)(Xl + tid * HR + i) = *(const float4*)(out_cur + (size_t)sn * DR + i);
      *(float4*)(Hl + tid * HR + i) = *(const float4*)(h + (size_t)ec * HR + i);
    }
  }

  v8f cacc[2][3];
  const v8f vz = {0.f, 0.f, 0.f, 0.f, 0.f, 0.f, 0.f, 0.f};
  #pragma unroll
  for (int mt = 0; mt < 2; ++mt)
    #pragma unroll
    for (int ot = 0; ot < 3; ++ot) cacc[mt][ot] = vz;

  for (int kb0 = 0; kb0 < KPAD; kb0 += KC) {
    __syncthreads();
    int rows = (KPAD - kb0) < KC ? (KPAD - kb0) : KC;
    // chunk is contiguous in the blocked layout too: copy rows*48 floats
    if (tid < rows) {
      #pragma unroll
      for (int o = 0; o < DR; o += 4)
        *(float4*)(Wl + tid * DR + o) = *(const float4*)(wm + (size_t)(kb0 + tid) * DR + o);
    }
    __syncthreads();

    int nkb = rows >> 2;
    for (int kb = 0; kb < nkb; ++kb) {
      int cg = kb0 + kb * 4 + 2 * lhalf;   // global c for this lane's vgpr 0
      int iq = cg / GJ;                    // uniform across the lane's k-pair
      int jq = cg - iq * GJ;               // always even
      // A fragments (16x4 f32 layout: vgpr d, lane half hf -> k = d + 2*hf)
      v2f a[2];
      #pragma unroll
      for (int mt = 0; mt < 2; ++mt) {
        int ew = wave * 32 + mt * 16 + lm;              // edge row within WG
        float xv = Xl[ew * HR + iq];                    // 1 x b32
        v2f  hv = *(const v2f*)(Hl + ew * HR + jq);     // 1 x b64 (8B aligned)
        a[mt][0] = xv * hv[0];
        a[mt][1] = xv * hv[1];
      }
      // B fragments: blocked layout -> two K-values adjacent, one b64 per o-tile
      const float* wb = Wl + kb * (DR * 4) + 2 * lhalf;
      v2f bfr[3];
      #pragma unroll
      for (int ot = 0; ot < 3; ++ot)
        bfr[ot] = *(const v2f*)(wb + (ot * 16 + lm) * 4);
      #pragma unroll
      for (int mt = 0; mt < 2; ++mt)
        #pragma unroll
        for (int ot = 0; ot < 3; ++ot)
          cacc[mt][ot] = __builtin_amdgcn_wmma_f32_16x16x4_f32(
              false, a[mt], false, bfr[ot], (short)0, cacc[mt][ot], false, false);
    }
  }

  // scatter: C layout vgpr r, lane half hf -> row m = r + 8*hf, col n = lane%16
  #pragma unroll
  for (int mt = 0; mt < 2; ++mt) {
    #pragma unroll
    for (int r = 0; r < 8; ++r) {
      int m  = r + 8 * lhalf;
      int ew = wave * 32 + mt * 16 + m;
      int eg = ebase + ew;
      if (eg < E) {
        int dn = Dl[ew];
        #pragma unroll
        for (int ot = 0; ot < 3; ++ot) {
          int o = ot * 16 + lm;
          if (o < DD) unsafeAtomicAdd(&neigh[(size_t)dn * DR + o], cacc[mt][ot][r]);
        }
      }
    }
  }
}

// out_nxt[n,o] = msg_b[o] + sum_i mw[o][i]*relu(neigh*inv + out + cb)[i] + mw[o][42+i]*out[i]
__global__ void k_node_update(const float* __restrict__ out_cur, const float* __restrict__ neigh,
                              const float* __restrict__ inv, const float* __restrict__ cb,
                              const float* __restrict__ mw, const float* __restrict__ mb,
                              float* __restrict__ out_nxt, int N) {
  int idx = blockIdx.x * blockDim.x + threadIdx.x;
  int n = idx / DR, o = idx % DR;
  if (n >= N) return;
  if (o >= DD) { out_nxt[n * DR + o] = 0.f; return; }
  float iv = inv[n];
  float s = mb[o];
  #pragma unroll
  for (int i = 0; i < DD; ++i) {
    float oc = out_cur[n * DR + i];
    float m  = fmaxf(neigh[n * DR + i] * iv + oc + cb[i], 0.f);
    s += mw[o * (2 * DD) + i] * m + mw[o * (2 * DD) + DD + i] * oc;
  }
  out_nxt[n * DR + o] = s;
}

__global__ void k_final_scatter(const float* __restrict__ out, const int* __restrict__ src,
                                const int* __restrict__ dst, float* __restrict__ acc, int E) {
  int idx = blockIdx.x * blockDim.x + threadIdx.x;
  int e = idx / DR, o = idx % DR;
  if (e >= E || o >= DD) return;
  unsafeAtomicAdd(&acc[(size_t)dst[e] * DR + o], out[(size_t)src[e] * DR + o]);
}

__global__ void k_final_proj(const float* __restrict__ out, const float* __restrict__ acc,
                             const float* __restrict__ inv, const float* __restrict__ pw,
                             const float* __restrict__ pb, const float* __restrict__ nf,
                             float* __restrict__ res, int N) {
  int idx = blockIdx.x * blockDim.x + threadIdx.x;
  int n = idx / DR, o = idx % DR;
  if (n >= N || o >= DD) return;
  float iv = inv[n];
  float s = pb[o];
  #pragma unroll
  for (int i = 0; i < DD; ++i)
    s += pw[o * (2 * DD) + i] * out[n * DR + i]
       + pw[o * (2 * DD) + DD + i] * (acc[n * DR + i] * iv);
  res[n * DD + o] = s + nf[n * DD + o];
}

extern "C" void kernel_launch(void* const* d_in, const int* in_sizes, int n_in,
                              void* d_out, int out_size, void* d_ws, size_t ws_size,
                              hipStream_t stream) {
  const float* n_feat = (const float*)d_in[0];
  const float* e_feat = (const float*)d_in[1];
  const int*   src    = (const int*)d_in[2];
  const int*   dst    = (const int*)d_in[3];
  const float* lin0_w = (const float*)d_in[4];
  const float* lin0_b = (const float*)d_in[5];
  const float* en1_w  = (const float*)d_in[6];
  const float* en1_b  = (const float*)d_in[7];
  const float* ln_g   = (const float*)d_in[8];
  const float* ln_b   = (const float*)d_in[9];
  const float* en2_w  = (const float*)d_in[10];
  const float* en2_b  = (const float*)d_in[11];
  const float* conv_b = (const float*)d_in[12];
  const float* msg_w  = (const float*)d_in[13];
  const float* msg_b  = (const float*)d_in[14];
  const float* proj_w = (const float*)d_in[15];
  const float* proj_b = (const float*)d_in[16];

  const int N = in_sizes[0] / DD;
  const int E = in_sizes[2];

  float* outA  = (float*)d_ws;
  float* outB  = outA  + (size_t)N * DR;
  float* hbuf  = outB  + (size_t)N * DR;
  float* wmat  = hbuf  + (size_t)E * HR;
  float* neigh = wmat  + (size_t)KPAD * DR;
  float* inv   = neigh + (size_t)N * DR;
  int*   cnt   = (int*)(inv + N);

  auto cdiv = [](int a, int b) { return (a + b - 1) / b; };

  k_zero<<<cdiv(N, 256), 256, 0, stream>>>((float*)cnt, N);
  k_indeg<<<cdiv(E, 256), 256, 0, stream>>>(dst, cnt, E);
  k_inv<<<cdiv(N, 256), 256, 0, stream>>>(cnt, inv, N);
  k_lin0<<<cdiv(N * DR, 256), 256, 0, stream>>>(n_feat, lin0_w, lin0_b, outA, N);
  k_edge_mlp<<<cdiv(E, 256), 256, 0, stream>>>(e_feat, en1_w, en1_b, ln_g, ln_b, hbuf, E);
  k_build_wmat<<<cdiv(KPAD * DR, 256), 256, 0, stream>>>(en2_w, en2_b, wmat);

  float* cur = outA;
  float* nxt = outB;
  for (int s = 0; s < 3; ++s) {
    k_zero<<<cdiv(N * DR, 256), 256, 0, stream>>>(neigh, N * DR);
    k_msg_gemm<<<cdiv(E, WGE), 256, 0, stream>>>(cur, hbuf, wmat, src, dst, neigh, E);
    k_node_update<<<cdiv(N * DR, 256), 256, 0, stream>>>(cur, neigh, inv, conv_b,
                                                         msg_w, msg_b, nxt, N);
    float* t = cur; cur = nxt; nxt = t;
  }
  k_zero<<<cdiv(N * DR, 256), 256, 0, stream>>>(neigh, N * DR);
  k_final_scatter<<<cdiv(E * DR, 256), 256, 0, stream>>>(cur, src, dst, neigh, E);
  k_final_proj<<<cdiv(N * DR, 256), 256, 0, stream>>>(cur, neigh, inv, proj_w, proj_b,
                                                      n_feat, (float*)d_out, N);
}